// MoEFFN_25640954757706
// MI455X (gfx1250) — compile-verified
//
#include <hip/hip_runtime.h>
#include <math.h>

// ---------------- problem constants ----------------
constexpr int Nn = 4096;   // B*T
constexpr int Dd = 1024;
constexpr int Hh = 2048;
constexpr int Ee = 16;     // real experts
constexpr int Kk = 4;      // top-k
constexpr int NE = 17;     // 16 routed + 1 shared

// ---------------- workspace layout (bytes) ----------------
constexpr size_t CNT_OFF  = 0;                                   // 17 ints
constexpr size_t NULL_OFF = 128;                                 // 1 int
constexpr size_t PART_OFF = 256;                                 // 16*17 floats
constexpr size_t IDX_OFF  = 4096;                                // NE*Nn ints
constexpr size_t WL_OFF   = IDX_OFF + (size_t)NE * Nn * 4;       // NE*Nn floats
constexpr size_t HBUF_OFF = WL_OFF + (size_t)NE * Nn * 4;        // Nn*Hh floats

typedef __attribute__((ext_vector_type(2))) float v2f;
typedef __attribute__((ext_vector_type(4))) float v4f;
typedef __attribute__((ext_vector_type(8))) float v8f;

// ---------------- CDNA5 async copy helpers ----------------
// 16B/lane global -> LDS, tracked by ASYNCcnt (no VGPR round-trip).
// LDS generic addresses carry the LDS byte offset in the low 32 bits.
__device__ __forceinline__ void async_ld16(float* lds, const float* g) {
  unsigned off = (unsigned)(unsigned long long)(void*)lds;
  asm volatile("global_load_async_to_lds_b128 %0, %1, off"
               :: "v"(off), "v"(g) : "memory");
}
__device__ __forceinline__ void wait_async0() {
  asm volatile("s_wait_asynccnt 0" ::: "memory");
}

// ---------------- init: zero out/accumulators, fill shared-expert list ----------------
__global__ void moe_init_kernel(float* __restrict__ out, int* __restrict__ cnt,
                                int* __restrict__ nullcnt, float* __restrict__ part,
                                int* __restrict__ idx16, float* __restrict__ w16) {
  const int gid = blockIdx.x * blockDim.x + threadIdx.x;
  const int stride = gridDim.x * blockDim.x;
  for (long i = gid; i < (long)Nn * Dd; i += stride) out[i] = 0.f;
  if (gid < Nn) { idx16[gid] = gid; w16[gid] = 1.f; }
  if (gid < NE) cnt[gid] = (gid == 16) ? Nn : 0;
  if (gid < 16 * 17) part[gid] = 0.f;
  if (gid == 0) *nullcnt = 0;
}

// ---------------- router: logits, softmax(32), top-4, gather lists, aux partials ----
__global__ __launch_bounds__(256)
void moe_router_kernel(const float* __restrict__ x, const float* __restrict__ gw,
                       const float* __restrict__ lbias, const float* __restrict__ nulll,
                       int* __restrict__ cnt, int* __restrict__ idxl, float* __restrict__ wl,
                       int* __restrict__ nullcnt, float* __restrict__ part) {
  const int n = blockIdx.x * 256 + threadIdx.x;
  float acc[16];
#pragma unroll
  for (int e = 0; e < 16; e++) acc[e] = 0.f;
  const float* xr = x + (long)n * Dd;
  for (int d = 0; d < Dd; d++) {
    const float xv = xr[d];
    const float* g = gw + d * 16;
#pragma unroll
    for (int e = 0; e < 16; e++) acc[e] += xv * g[e];
  }
#pragma unroll
  for (int e = 0; e < 16; e++) acc[e] += lbias[e];
  const float L = nulll[0];

  // softmax over real logits only (for P_real)
  float mr = acc[0];
#pragma unroll
  for (int e = 1; e < 16; e++) mr = fmaxf(mr, acc[e]);
  float p16[16];
  float s16 = 0.f;
#pragma unroll
  for (int e = 0; e < 16; e++) { p16[e] = expf(acc[e] - mr); s16 += p16[e]; }

  // softmax over 32 logits (16 real + 16 copies of null)
  const float m = fmaxf(mr, L);
  float s = 16.f * expf(L - m);
#pragma unroll
  for (int e = 0; e < 16; e++) s += expf(acc[e] - m);
  const float lse = m + logf(s);

  // top-4: null copies all share prob exp(L-m)/s; real beats null on ties (lower index)
  unsigned usedMask = 0;
  int   pe[4]; float pw[4];
  int ns = 0; float S = 0.f;
#pragma unroll
  for (int kq = 0; kq < 4; kq++) {
    float best = -3.4e38f; int bi = 0;
#pragma unroll
    for (int e = 0; e < 16; e++)
      if (!((usedMask >> e) & 1u) && acc[e] > best) { best = acc[e]; bi = e; }
    usedMask |= (1u << bi);
    if (best >= L) { pe[ns] = bi; pw[ns] = expf(best - m) / s; S += pw[ns]; ns++; }
  }
  const float inv = 1.f / fmaxf(S, 1e-6f);
  for (int q = 0; q < ns; q++) {
    const int e = pe[q];
    const int pos = atomicAdd(&cnt[e], 1);     // int atomic: deterministic counts
    idxl[e * Nn + pos] = n;
    wl[e * Nn + pos] = pw[q] * inv;
  }
  atomicAdd(nullcnt, Kk - ns);

  // deterministic per-block reduction of float partials (16 x sumP + lse^2)
  __shared__ float red[256];
  for (int c = 0; c < 17; c++) {
    red[threadIdx.x] = (c < 16) ? (p16[c] / s16) : (lse * lse);
    __syncthreads();
    for (int off = 128; off > 0; off >>= 1) {
      if (threadIdx.x < off) red[threadIdx.x] += red[threadIdx.x + off];
      __syncthreads();
    }
    if (threadIdx.x == 0) part[blockIdx.x * 17 + c] = red[0];
    __syncthreads();
  }
}

// ---------------- aux-loss finalize (fixed order => deterministic) ----------------
__global__ void moe_finalize_kernel(const int* __restrict__ cnt, const int* __restrict__ nullcnt,
                                    const float* __restrict__ part, float* __restrict__ out_aux) {
  if (threadIdx.x != 0 || blockIdx.x != 0) return;
  float sumP[16]; float lse2 = 0.f;
  for (int e = 0; e < 16; e++) sumP[e] = 0.f;
  for (int b = 0; b < 16; b++) {
    for (int e = 0; e < 16; e++) sumP[e] += part[b * 17 + e];
    lse2 += part[b * 17 + 16];
  }
  float tot = 0.f;
  for (int e = 0; e < 16; e++) tot += (float)cnt[e];
  const float invtot = 1.f / fmaxf(tot, 1e-6f);
  float Lbal = 0.f;
  for (int e = 0; e < 16; e++) Lbal += ((float)cnt[e] * invtot) * (sumP[e] / (float)Nn);
  Lbal *= 16.f;
  const float nr = (float)(*nullcnt) / (float)(Nn * Kk);
  const float Lnull = (nr - 0.5f) * (nr - 0.5f);
  const float Lz = lse2 / (float)Nn;
  *out_aux = 0.02f * Lbal + 0.001f * Lz + 0.01f * Lnull;
}

// ---------------- WMMA helper ----------------
__device__ __forceinline__ v8f wmma_f32(v2f a, v2f b, v8f c) {
  return __builtin_amdgcn_wmma_f32_16x16x4_f32(false, a, false, b, (short)0, c, false, false);
}

// ---------------- GEMM1: h = silu(X*Wg) * (X*Wu), gathered rows ----------------
// 64x64 block tile, BK=16, 8 waves each owning 16(M)x32(N) of C for both matrices.
// Double-buffered LDS; B tiles streamed with async global->LDS copies.
__global__ __launch_bounds__(256)
void moe_gemm1_kernel(const float* __restrict__ X, const float* __restrict__ Wg,
                      const float* __restrict__ Wu, const int* __restrict__ idxl,
                      const int* __restrict__ cntPtr, float* __restrict__ hbuf) {
  const int cnt = *cntPtr;
  const int rowBase = blockIdx.x * 64;
  if (rowBase >= cnt) return;                 // uniform early-exit (EXEC stays all-1s)
  const int colBase = blockIdx.y * 64;

  __shared__ float As[2][64 * 17];            // padded stride 17: conflict-free frag reads
  __shared__ float Bg[2][16 * 64];
  __shared__ float Bu[2][16 * 64];

  const int tid = threadIdx.x;
  const int lane = tid & 31;
  const int wave = tid >> 5;                  // 8 waves (wave32)
  const int wm = wave & 3;                    // M tile 0..3
  const int wn = wave >> 2;                   // N group 0..1 (32 cols)

  const int ar = tid >> 2;                    // A: 4 threads/row, 64 rows
  const int ac = (tid & 3) * 4;
  const int br = tid >> 4;                    // B: 16 threads/row, 16 rows
  const int bc = (tid & 15) * 4;

  const int rgA = rowBase + ar;
  const int tok = idxl[(rgA < cnt) ? rgA : rowBase];

  const int mloc = lane & 15;
  const int laneHi = lane >> 4;
  const int khalf = laneHi * 2;               // K offset 0 or 2 within the 4-wide step

  v8f cg0 = {0}, cg1 = {0}, cu0 = {0}, cu1 = {0};

  // stage 0 preload
  {
    v4f av = *(const v4f*)(X + (long)tok * Dd + ac);
#pragma unroll
    for (int j = 0; j < 4; j++) As[0][ar * 17 + ac + j] = av[j];
    async_ld16(&Bg[0][br * 64 + bc], Wg + (long)br * Hh + colBase + bc);
    async_ld16(&Bu[0][br * 64 + bc], Wu + (long)br * Hh + colBase + bc);
  }
  wait_async0();
  __syncthreads();

  for (int kb = 0; kb < Dd; kb += 16) {
    const int cur = (kb >> 4) & 1;
    const int nxt = cur ^ 1;
    if (kb + 16 < Dd) {                       // stream next stage while computing
      v4f av = *(const v4f*)(X + (long)tok * Dd + (kb + 16) + ac);
#pragma unroll
      for (int j = 0; j < 4; j++) As[nxt][ar * 17 + ac + j] = av[j];
      async_ld16(&Bg[nxt][br * 64 + bc], Wg + (long)(kb + 16 + br) * Hh + colBase + bc);
      async_ld16(&Bu[nxt][br * 64 + bc], Wu + (long)(kb + 16 + br) * Hh + colBase + bc);
    }
#pragma unroll
    for (int kk = 0; kk < 16; kk += 4) {
      const int k0 = kk + khalf;
      v2f a;
      a.x = As[cur][(wm * 16 + mloc) * 17 + k0];
      a.y = As[cur][(wm * 16 + mloc) * 17 + k0 + 1];
      const int n0 = wn * 32 + mloc;
      v2f bg0 = { Bg[cur][k0 * 64 + n0],      Bg[cur][(k0 + 1) * 64 + n0]      };
      v2f bg1 = { Bg[cur][k0 * 64 + n0 + 16], Bg[cur][(k0 + 1) * 64 + n0 + 16] };
      v2f bu0 = { Bu[cur][k0 * 64 + n0],      Bu[cur][(k0 + 1) * 64 + n0]      };
      v2f bu1 = { Bu[cur][k0 * 64 + n0 + 16], Bu[cur][(k0 + 1) * 64 + n0 + 16] };
      cg0 = wmma_f32(a, bg0, cg0);
      cg1 = wmma_f32(a, bg1, cg1);
      cu0 = wmma_f32(a, bu0, cu0);
      cu1 = wmma_f32(a, bu1, cu1);
    }
    wait_async0();                            // own-wave async copies done
    __syncthreads();                          // all waves: reads done + buffers visible
  }

  // epilogue: silu(g)*u -> compact hbuf rows
#pragma unroll
  for (int vi = 0; vi < 8; vi++) {
    const int rloc = wm * 16 + laneHi * 8 + vi;
    const int rg = rowBase + rloc;
    if (rg < cnt) {
      const int c0 = colBase + wn * 32 + mloc;
      const float g0 = cg0[vi], u0 = cu0[vi];
      const float g1 = cg1[vi], u1 = cu1[vi];
      hbuf[(long)rg * Hh + c0]      = (g0 / (1.f + expf(-g0))) * u0;
      hbuf[(long)rg * Hh + c0 + 16] = (g1 / (1.f + expf(-g1))) * u1;
    }
  }
}

// ---------------- GEMM2: out[token] += w * (h * Wd), scatter accumulate ----------------
__global__ __launch_bounds__(256)
void moe_gemm2_kernel(const float* __restrict__ hbuf, const float* __restrict__ Wd,
                      const int* __restrict__ idxl, const float* __restrict__ wlst,
                      const int* __restrict__ cntPtr, float* __restrict__ out) {
  const int cnt = *cntPtr;
  const int rowBase = blockIdx.x * 64;
  if (rowBase >= cnt) return;
  const int colBase = blockIdx.y * 64;        // D columns

  __shared__ float As[2][64 * 17];
  __shared__ float Bs[2][16 * 64];

  const int tid = threadIdx.x;
  const int lane = tid & 31;
  const int wave = tid >> 5;
  const int wm = wave & 3;
  const int wn = wave >> 2;

  const int ar = tid >> 2;
  const int ac = (tid & 3) * 4;
  const int br = tid >> 4;
  const int bc = (tid & 15) * 4;

  const int rgA = rowBase + ar;
  const int rsrc = (rgA < cnt) ? rgA : rowBase;   // compact hbuf rows

  const int mloc = lane & 15;
  const int laneHi = lane >> 4;
  const int khalf = laneHi * 2;

  v8f c0 = {0}, c1 = {0};

  {
    v4f av = *(const v4f*)(hbuf + (long)rsrc * Hh + ac);
#pragma unroll
    for (int j = 0; j < 4; j++) As[0][ar * 17 + ac + j] = av[j];
    async_ld16(&Bs[0][br * 64 + bc], Wd + (long)br * Dd + colBase + bc);
  }
  wait_async0();
  __syncthreads();

  for (int kb = 0; kb < Hh; kb += 16) {
    const int cur = (kb >> 4) & 1;
    const int nxt = cur ^ 1;
    if (kb + 16 < Hh) {
      v4f av = *(const v4f*)(hbuf + (long)rsrc * Hh + (kb + 16) + ac);
#pragma unroll
      for (int j = 0; j < 4; j++) As[nxt][ar * 17 + ac + j] = av[j];
      async_ld16(&Bs[nxt][br * 64 + bc], Wd + (long)(kb + 16 + br) * Dd + colBase + bc);
    }
#pragma unroll
    for (int kk = 0; kk < 16; kk += 4) {
      const int k0 = kk + khalf;
      v2f a;
      a.x = As[cur][(wm * 16 + mloc) * 17 + k0];
      a.y = As[cur][(wm * 16 + mloc) * 17 + k0 + 1];
      const int n0 = wn * 32 + mloc;
      v2f b0 = { Bs[cur][k0 * 64 + n0],      Bs[cur][(k0 + 1) * 64 + n0]      };
      v2f b1 = { Bs[cur][k0 * 64 + n0 + 16], Bs[cur][(k0 + 1) * 64 + n0 + 16] };
      c0 = wmma_f32(a, b0, c0);
      c1 = wmma_f32(a, b1, c1);
    }
    wait_async0();
    __syncthreads();
  }

#pragma unroll
  for (int vi = 0; vi < 8; vi++) {
    const int rloc = wm * 16 + laneHi * 8 + vi;
    const int rg = rowBase + rloc;
    if (rg < cnt) {
      const int tok = idxl[rg];
      const float w = wlst[rg];
      const int cc = colBase + wn * 32 + mloc;
      float* o = out + (long)tok * Dd + cc;
      o[0]  += w * c0[vi];
      o[16] += w * c1[vi];
    }
  }
}

// ---------------- host entry ----------------
extern "C" void kernel_launch(void* const* d_in, const int* in_sizes, int n_in,
                              void* d_out, int out_size, void* d_ws, size_t ws_size,
                              hipStream_t stream) {
  const float* x          = (const float*)d_in[0];
  const float* gate_w     = (const float*)d_in[1];
  const float* logit_bias = (const float*)d_in[2];
  const float* null_logit = (const float*)d_in[3];
  const float* W_gate     = (const float*)d_in[4];
  const float* W_up       = (const float*)d_in[5];
  const float* W_down     = (const float*)d_in[6];
  const float* ws_gate    = (const float*)d_in[7];
  const float* ws_up      = (const float*)d_in[8];
  const float* ws_down    = (const float*)d_in[9];
  float* out = (float*)d_out;

  char* ws = (char*)d_ws;
  int*   cnt     = (int*)(ws + CNT_OFF);
  int*   nullcnt = (int*)(ws + NULL_OFF);
  float* part    = (float*)(ws + PART_OFF);
  int*   idxl    = (int*)(ws + IDX_OFF);
  float* wl      = (float*)(ws + WL_OFF);
  float* hbuf    = (float*)(ws + HBUF_OFF);

  moe_init_kernel<<<1024, 256, 0, stream>>>(out, cnt, nullcnt, part,
                                            idxl + 16 * Nn, wl + 16 * Nn);
  moe_router_kernel<<<16, 256, 0, stream>>>(x, gate_w, logit_bias, null_logit,
                                            cnt, idxl, wl, nullcnt, part);
  moe_finalize_kernel<<<1, 32, 0, stream>>>(cnt, nullcnt, part, out + (long)Nn * Dd);

  for (int e = 0; e < NE; e++) {
    const float* Wg = (e < 16) ? (W_gate + (long)e * Dd * Hh) : ws_gate;
    const float* Wu = (e < 16) ? (W_up   + (long)e * Dd * Hh) : ws_up;
    const float* Wd = (e < 16) ? (W_down + (long)e * Hh * Dd) : ws_down;
    moe_gemm1_kernel<<<dim3(Nn / 64, Hh / 64), 256, 0, stream>>>(
        x, Wg, Wu, idxl + (long)e * Nn, cnt + e, hbuf);
    moe_gemm2_kernel<<<dim3(Nn / 64, Dd / 64), 256, 0, stream>>>(
        hbuf, Wd, idxl + (long)e * Nn, wl + (long)e * Nn, cnt + e, out);
  }
}